// Dconv_46170898432133
// MI455X (gfx1250) — compile-verified
//
#include <hip/hip_runtime.h>

// gfx1250 / CDNA5: wave32, WMMA f32 16x16x4.
typedef __attribute__((ext_vector_type(2))) float v2f;
typedef __attribute__((ext_vector_type(8))) float v8f;

#define BATCH   8
#define CHANS   32
#define HIN     224
#define WIN     224
#define HO      222
#define WO      222
#define NCONV   8
#define TILES_W 14            // ceil(222/16)
#define ROWS_PER_BLOCK 8      // 8 waves per 256-thread block, 1 row per wave

__global__ __launch_bounds__(256)
void dconv_wmma_f32(const float* __restrict__ x,
                    const float* __restrict__ w,    // [8,3,3] flat
                    const float* __restrict__ bv,   // [8]
                    float* __restrict__ out)        // [8, 8*32, 222, 222]
{
    const int lane = threadIdx.x & 31;
    const int wvid = threadIdx.x >> 5;
    const int row  = blockIdx.x * ROWS_PER_BLOCK + wvid;
    if (row >= HO) return;                 // uniform per wave

    const int plane = blockIdx.y;          // bat*CHANS + cch
    const int bat   = plane >> 5;
    const int cch   = plane & 31;

    const int n  = lane & 15;              // pixel within tile / D column
    const int hi = lane >> 4;              // K-half select for A/B operands
    const int h2 = 2 * hi;

    // ---- A operand (weights), loaded once per wave ----
    // ISA 16x4 f32 layout: VGPR0 = K=4t+0 (lanes 0-15) / K=4t+2 (lanes 16-31)
    //                      VGPR1 = K=4t+1 (lanes 0-15) / K=4t+3 (lanes 16-31)
    // D rows m>=8 are discarded, so A rows 8-15 need no masking (clamped idx).
    const int mC = (n < NCONV) ? n : (NCONV - 1);
    const float* wr = w + mC * 9;
    v2f A0, A1, A2;
    A0.x = wr[0 + h2];                     // taps 0 / 2
    A0.y = wr[1 + h2];                     // taps 1 / 3
    A1.x = wr[4 + h2];                     // taps 4 / 6
    A1.y = wr[5 + h2];                     // taps 5 / 7
    const float w8 = wr[8];                // unconditional load (no branch)
    A2.x = hi ? 0.0f : w8;                 // tap 8 (lo); tap 10 must be 0
    A2.y = 0.0f;                           // taps 9/11 zero => B column free

    // ---- bias (uniform, one scalar block load) ----
    float bvr[NCONV];
#pragma unroll
    for (int j = 0; j < NCONV; ++j) bvr[j] = bv[j];

    // ---- B-operand addressing ----
    // taps per WMMA: t0 lo (0,0),(0,1) | hi (0,2),(1,0)
    //                t1 lo (1,1),(1,2) | hi (2,0),(2,1)
    //                t2 lo (2,2)       | hi zero-weighted
    // Middle three loads differ between halves by exactly WIN-1 = 223.
    const int hb = hi * (WIN - 1);         // 0 or 223

    const float* xrow = x + plane * (HIN * WIN) + row * WIN;   // uniform base

    const unsigned hw      = (unsigned)HO * WO;
    const unsigned jstride = (unsigned)CHANS * hw;             // 1,577,088
    float* ob = out + ((unsigned)(bat * NCONV) * CHANS + cch) * hw
                    + (unsigned)row * WO;                      // uniform base

    auto tileAcc = [&](int ccol) -> v8f {
        const int c1 = ccol + hb;
        v2f B0, B1, B2;
        B0.x = xrow[ccol + h2];            // (0,0) / (0,2)
        B0.y = xrow[c1 + 1];               // (0,1) / (1,0)
        B1.x = xrow[c1 + WIN + 1];         // (1,1) / (2,0)
        B1.y = xrow[c1 + WIN + 2];         // (1,2) / (2,1)
        B2.x = xrow[ccol + 2 * WIN + 2];   // (2,2); hi half hits A[*,10]==0
        B2.y = B2.x;                       // A[*,11]==0: value irrelevant

        v8f acc = {};
        acc = __builtin_amdgcn_wmma_f32_16x16x4_f32(
            false, A0, false, B0, (short)0, acc, false, false);
        acc = __builtin_amdgcn_wmma_f32_16x16x4_f32(
            false, A1, false, B1, (short)0, acc, false, false);
        acc = __builtin_amdgcn_wmma_f32_16x16x4_f32(
            false, A2, false, B2, (short)0, acc, false, false);
        return acc;
    };

    // ---- main tiles: cols <= 207+450 in-bounds, no clamping needed ----
    for (int tile = 0; tile < TILES_W - 1; ++tile) {
        const int col = tile * 16 + n;
        v8f acc = tileAcc(col);
        // D: VGPR j, lanes 0-15 -> kernel j, pixel col. NT stores: 404 MB
        // write-once stream; keep L2 for the reused input.
        if (hi == 0) {
#pragma unroll
            for (int j = 0; j < NCONV; ++j) {
                __builtin_nontemporal_store(acc[j] + bvr[j],
                                            ob + j * jstride + col);
            }
        }
    }

    // ---- epilogue tile: cols 208..223, clamp loads, predicate stores ----
    {
        const int col  = (TILES_W - 1) * 16 + n;
        const int ccol = (col < WO) ? col : (WO - 1);
        v8f acc = tileAcc(ccol);
        if (hi == 0 && col < WO) {
#pragma unroll
            for (int j = 0; j < NCONV; ++j) {
                __builtin_nontemporal_store(acc[j] + bvr[j],
                                            ob + j * jstride + col);
            }
        }
    }
}

extern "C" void kernel_launch(void* const* d_in, const int* in_sizes, int n_in,
                              void* d_out, int out_size, void* d_ws, size_t ws_size,
                              hipStream_t stream) {
    const float* x  = (const float*)d_in[0];   // [8,32,224,224]
    const float* w  = (const float*)d_in[1];   // [1,8,3,3]
    const float* bv = (const float*)d_in[2];   // [8]
    float* out = (float*)d_out;                // [8, 256, 222, 222]

    dim3 grid((HO + ROWS_PER_BLOCK - 1) / ROWS_PER_BLOCK,  // 28
              BATCH * CHANS);                              // 256 planes
    dconv_wmma_f32<<<grid, 256, 0, stream>>>(x, w, bv, out);
}